// QuantumNeuralNetwork_32461362823963
// MI455X (gfx1250) — compile-verified
//
#include <hip/hip_runtime.h>
#include <math.h>

typedef __attribute__((ext_vector_type(2))) float v2f;
typedef __attribute__((ext_vector_type(8))) float v8f;

#define NQ   16
#define DIM  65536
#define BATCH 128
#define NL   8
#define NOBS 5

// Composed CNOT-ring permutation: new bit at position p (p=15-q) is the
// suffix-XOR of bits p..15, except bit15 which is XOR of bits 0..14.
__device__ __forceinline__ unsigned cnot_perm(unsigned e) {
    unsigned t = e;
    t ^= t >> 1; t ^= t >> 2; t ^= t >> 4; t ^= t >> 8;
    t &= 0xFFFFu;
    unsigned b15 = (t ^ (e >> 15)) & 1u;
    return (t & 0x7FFFu) | (b15 << 15);
}

// U[((l*2+h)*256+i)*256+k] = prod over 8 qubits of RY(theta) 2x2 entries.
// Qubit (h*8 + (7-p)) maps to bit p of the 8-bit sub-index (qubit 0 = MSB).
__global__ void build_unitaries(const float* __restrict__ thetas,
                                float* __restrict__ U) {
    unsigned g = blockIdx.x * blockDim.x + threadIdx.x; // 8*2*256*256 threads
    unsigned k = g & 255u;
    unsigned i = (g >> 8) & 255u;
    unsigned h = (g >> 16) & 1u;
    unsigned l = g >> 17;
    float prod = 1.0f;
#pragma unroll
    for (int p = 0; p < 8; ++p) {
        int q = (int)h * 8 + (7 - p);
        float th = 0.5f * thetas[l * NQ + q];
        float s, c;
        __sincosf(th, &s, &c);
        int r = (i >> p) & 1;
        int t = (k >> p) & 1;
        // RY = [[c, -s], [s, c]]
        float f = r ? (t ? c : s) : (t ? -s : c);
        prod *= f;
    }
    U[g] = prod;
}

// One layer: S' = A * S * B^T (real & imag), then scatter via cnot_perm.
// Block = 256 threads = 8 waves; computes a 32-row slab of the 256x256 state.
// Jobs are (row-tile, col-tile); each job fuses re+im so A/B fragments are
// loaded once and feed two WMMAs.
__global__ __launch_bounds__(256)
void layer_kernel(const float* __restrict__ A, const float* __restrict__ Bm,
                  const float* __restrict__ Sr, const float* __restrict__ Si,
                  float* __restrict__ Or, float* __restrict__ Oi) {
    __shared__ float Tlds[2][32][256];   // 64 KB: T = A_slab * S (re, im)

    const int slab  = blockIdx.x;        // 0..7  (32 rows each)
    const int batch = blockIdx.y;        // 0..127
    const int lane  = threadIdx.x & 31;
    const int wave  = threadIdx.x >> 5;
    const int n16   = lane & 15;
    const int khalf = (lane >> 4) * 2;   // K offset {0,2} per A/B frag layout
    const int mofs  = (lane >> 4) * 8;   // C/D row offset for upper lanes
    const size_t soff = (size_t)batch * DIM;
    const int rowBase = slab * 32;

    // ---- Phase 1: T = A[rows,:] * S  (32 tile jobs over 8 waves) ----
    for (int t = 0; t < 4; ++t) {
        int job = wave + t * 8;          // 0..31
        int ct = job & 15;               // col tile
        int rt = job >> 4;               // row tile (0..1)
        int row0 = rowBase + rt * 16;
        int col0 = ct * 16;
        size_t aBase = (size_t)(row0 + n16) * 256 + khalf;
        size_t sCol  = soff + col0 + n16;
        v8f accR = {};
        v8f accI = {};
#pragma unroll 4
        for (int k = 0; k < 256; k += 4) {
            v2f a;
            a.x = A[aBase + k];
            a.y = A[aBase + k + 1];
            size_t s0 = sCol + (size_t)(k + khalf) * 256;
            v2f br, bi;
            br.x = Sr[s0];
            br.y = Sr[s0 + 256];
            bi.x = Si[s0];
            bi.y = Si[s0 + 256];
            accR = __builtin_amdgcn_wmma_f32_16x16x4_f32(
                false, a, false, br, (short)0, accR, false, false);
            accI = __builtin_amdgcn_wmma_f32_16x16x4_f32(
                false, a, false, bi, (short)0, accI, false, false);
        }
#pragma unroll
        for (int v = 0; v < 8; ++v) {
            Tlds[0][rt * 16 + v + mofs][col0 + n16] = accR[v];
            Tlds[1][rt * 16 + v + mofs][col0 + n16] = accI[v];
        }
    }
    __syncthreads();

    // ---- Phase 2: S'_slab = T * B^T, scatter through CNOT permutation ----
    for (int t = 0; t < 4; ++t) {
        int job = wave + t * 8;
        int ct = job & 15;
        int rt = job >> 4;
        int col0 = ct * 16;
        size_t bBase = (size_t)(col0 + n16) * 256 + khalf;
        int trow = rt * 16 + n16;
        v8f accR = {};
        v8f accI = {};
#pragma unroll 4
        for (int l = 0; l < 256; l += 4) {
            v2f b;
            b.x = Bm[bBase + l];
            b.y = Bm[bBase + l + 1];
            v2f ar, ai;
            ar.x = Tlds[0][trow][l + khalf];
            ar.y = Tlds[0][trow][l + khalf + 1];
            ai.x = Tlds[1][trow][l + khalf];
            ai.y = Tlds[1][trow][l + khalf + 1];
            accR = __builtin_amdgcn_wmma_f32_16x16x4_f32(
                false, ar, false, b, (short)0, accR, false, false);
            accI = __builtin_amdgcn_wmma_f32_16x16x4_f32(
                false, ai, false, b, (short)0, accI, false, false);
        }
#pragma unroll
        for (int v = 0; v < 8; ++v) {
            unsigned i = (unsigned)(rowBase + rt * 16 + v + mofs);
            unsigned j = (unsigned)(col0 + n16);
            size_t n = soff + cnot_perm((i << 8) | j);
            Or[n] = accR[v];
            Oi[n] = accI[v];
        }
    }
}

// Per-batch: probs = r^2+i^2; logits[w] = sum probs * (+1/-1 by bit 15-w).
__global__ __launch_bounds__(256)
void observables_kernel(const float* __restrict__ Sr,
                        const float* __restrict__ Si,
                        float* __restrict__ out) {
    int b = blockIdx.x;
    int tid = threadIdx.x;
    size_t base = (size_t)b * DIM + (size_t)tid * 256;
    float s = 0.f;
#pragma unroll 4
    for (int m = 0; m < 256; ++m) {
        float x = Sr[base + m], y = Si[base + m];
        s += x * x + y * y;
    }
    // bit (15-w) of global index == bit (7-w) of tid (chunk is 256 contiguous)
    __shared__ float red[NOBS][256];
#pragma unroll
    for (int w = 0; w < NOBS; ++w) {
        float sign = ((tid >> (7 - w)) & 1) ? -1.f : 1.f;
        red[w][tid] = sign * s;
    }
    __syncthreads();
    for (int stride = 128; stride > 0; stride >>= 1) {
        if (tid < stride) {
#pragma unroll
            for (int w = 0; w < NOBS; ++w)
                red[w][tid] += red[w][tid + stride];
        }
        __syncthreads();
    }
    if (tid < NOBS) out[1 + b * NOBS + tid] = red[tid][0];
}

// loss = mean over batch of (logsumexp(logits) - logits[y])
__global__ void loss_kernel(const float* __restrict__ logits_in,
                            const int* __restrict__ y,
                            float* __restrict__ out) {
    __shared__ float red[128];
    int b = threadIdx.x;
    const float* lg = logits_in + 1 + b * NOBS;
    float mx = lg[0];
#pragma unroll
    for (int w = 1; w < NOBS; ++w) mx = fmaxf(mx, lg[w]);
    float se = 0.f;
#pragma unroll
    for (int w = 0; w < NOBS; ++w) se += expf(lg[w] - mx);
    float lse = mx + logf(se);
    int lbl = y[b];
    red[b] = lse - lg[lbl];
    __syncthreads();
    for (int stride = 64; stride > 0; stride >>= 1) {
        if (b < stride) red[b] += red[b + stride];
        __syncthreads();
    }
    if (b == 0) out[0] = red[0] * (1.f / 128.f);
}

extern "C" void kernel_launch(void* const* d_in, const int* in_sizes, int n_in,
                              void* d_out, int out_size, void* d_ws, size_t ws_size,
                              hipStream_t stream) {
    const float* zr     = (const float*)d_in[0];
    const float* zi     = (const float*)d_in[1];
    const float* thetas = (const float*)d_in[2];
    const int*   y      = (const int*)d_in[3];
    float* out = (float*)d_out;
    float* ws  = (float*)d_ws;

    // Workspace layout (floats): U (8*2*256*256 = 4MB), then 4 state planes
    // (2 buffers x re/im, each BATCH*DIM = 32MB). Total ~132MB, L2-resident.
    float* U = ws;
    size_t uoff  = (size_t)NL * 2 * 256 * 256;
    size_t plane = (size_t)BATCH * DIM;
    float* b0r = ws + uoff;
    float* b0i = b0r + plane;
    float* b1r = b0i + plane;
    float* b1i = b1r + plane;

    build_unitaries<<<(NL * 2 * 256 * 256) / 256, 256, 0, stream>>>(thetas, U);

    const float* inr = zr;
    const float* ini = zi;
    for (int l = 0; l < NL; ++l) {
        float* outr = (l & 1) ? b1r : b0r;
        float* outi = (l & 1) ? b1i : b0i;
        const float* A  = U + (size_t)(l * 2 + 0) * 256 * 256;
        const float* Bm = U + (size_t)(l * 2 + 1) * 256 * 256;
        layer_kernel<<<dim3(8, BATCH), 256, 0, stream>>>(A, Bm, inr, ini, outr, outi);
        inr = outr; ini = outi;
    }

    observables_kernel<<<BATCH, 256, 0, stream>>>(inr, ini, out);
    loss_kernel<<<1, 128, 0, stream>>>(out, y, out);
}